// FlowNet3D_12146167513434
// MI455X (gfx1250) — compile-verified
//
#include <hip/hip_runtime.h>

typedef float v2f __attribute__((ext_vector_type(2)));
typedef float v8f __attribute__((ext_vector_type(8)));

static constexpr int BB = 16;      // batch
static constexpr float BN_EPS = 1e-5f;

// ---------------------------------------------------------------------------
// fp32 GEMM via V_WMMA_F32_16X16X4_F32.
// Y[b] (MxN, row-major) = alpha * A(b) (MxK) * X[b] (KxN, row-major)
//   TRANSA==0: Aelem(m,k) = A[b*strideA + m*lda + k]   (lda==K for weights)
//   TRANSA==1: Aelem(m,k) = A[b*strideA + k*lda + m]   (attention: f1n^T)
// Each wave computes a 16x32 tile (2 accumulators sharing the A fragment);
// 4 waves per block cover 128 columns. GUARD=false path is fully branch-free.
// ISA layouts (CDNA5 ISA):
//   A 16x4 : lanes 0-15 row M=lane, v0:K=k0+0 v1:K=k0+1 ; lanes 16-31 K=k0+2/3
//   B 4x16 : lanes 0-15 col N=lane, v0:K=k0+0 v1:K=k0+1 ; lanes 16-31 K=k0+2/3
//   C 16x16: vgpr i -> row m0+i (lanes 0-15) / m0+8+i (lanes 16-31), col=lane&15
// ---------------------------------------------------------------------------
template<bool GUARD, bool TRANSA>
__global__ void k_gemm(const float* __restrict__ A, const float* __restrict__ X,
                       float* __restrict__ Y,
                       int M, int K, int N, int lda,
                       long strideA, long strideX, long strideY, float alpha)
{
    const int lane = threadIdx.x & 31;
    const int wv   = threadIdx.x >> 5;
    const int n0 = (blockIdx.x * 4 + wv) * 32;
    const int m0 = blockIdx.y * 16;
    const int b  = blockIdx.z;
    if (n0 >= N) return;                    // wave-uniform exit
    const int half = lane >> 4;
    const int l15  = lane & 15;
    const int mrow = m0 + l15;
    const int nc0  = n0 + l15;
    const int nc1  = nc0 + 16;
    const float* Ab = A + (long)b * strideA;
    const float* Xb = X + (long)b * strideX;
    float* Yb = Y + (long)b * strideY;

    v8f acc0 = {0.f, 0.f, 0.f, 0.f, 0.f, 0.f, 0.f, 0.f};
    v8f acc1 = {0.f, 0.f, 0.f, 0.f, 0.f, 0.f, 0.f, 0.f};

    if (!GUARD) {
        // ---- branch-free fast path: M%16==0, K%4==0, N%32==0 ----
        const float* Ap = TRANSA ? (Ab + (long)(half * 2) * lda + mrow)
                                 : (Ab + (long)mrow * lda + half * 2);
        const float* Bp = Xb + (long)(half * 2) * N + nc0;
        const long aoff1 = TRANSA ? (long)lda : 1L;
        const long astep = TRANSA ? 4L * lda : 4L;
        const long bstep = 4L * (long)N;
        const int iters = K >> 2;
#pragma unroll 2
        for (int i = 0; i < iters; ++i) {
            v2f av  = { Ap[0],      Ap[aoff1]      };
            v2f bv0 = { Bp[0],      Bp[N]          };
            v2f bv1 = { Bp[16],     Bp[N + 16]     };
            acc0 = __builtin_amdgcn_wmma_f32_16x16x4_f32(false, av, false, bv0,
                                                         (short)0, acc0, false, false);
            acc1 = __builtin_amdgcn_wmma_f32_16x16x4_f32(false, av, false, bv1,
                                                         (short)0, acc1, false, false);
            Ap += astep;
            Bp += bstep;
        }
        const int mbase = m0 + half * 8;
#pragma unroll
        for (int i = 0; i < 8; ++i) {
            long r = (long)(mbase + i) * N;
            Yb[r + nc0] = alpha * acc0[i];
            Yb[r + nc1] = alpha * acc1[i];
        }
    } else {
        // ---- guarded path: clamped addresses + selects (no divergent loads) ----
        const bool mok  = mrow < M;
        const int  mc   = mok ? mrow : 0;
        const bool n0ok = nc0 < N;
        const bool n1ok = nc1 < N;
        const int  nb0  = n0ok ? nc0 : 0;
        const int  nb1  = n1ok ? nc1 : 0;
        for (int k0 = 0; k0 < K; k0 += 4) {
            int ka  = k0 + half * 2;
            int ka0 = (ka     < K) ? ka     : K - 1;
            int ka1 = (ka + 1 < K) ? ka + 1 : K - 1;
            float a0, a1;
            if (TRANSA) {
                a0 = Ab[(long)ka0 * lda + mc];
                a1 = Ab[(long)ka1 * lda + mc];
            } else {
                a0 = Ab[(long)mc * lda + ka0];
                a1 = Ab[(long)mc * lda + ka1];
            }
            // out-of-range K must contribute zero: zero the A side only
            a0 = (ka     < K) ? a0 : 0.f;
            a1 = (ka + 1 < K) ? a1 : 0.f;
            v2f av  = { a0, a1 };
            v2f bv0 = { Xb[(long)ka0 * N + nb0], Xb[(long)ka1 * N + nb0] };
            v2f bv1 = { Xb[(long)ka0 * N + nb1], Xb[(long)ka1 * N + nb1] };
            acc0 = __builtin_amdgcn_wmma_f32_16x16x4_f32(false, av, false, bv0,
                                                         (short)0, acc0, false, false);
            acc1 = __builtin_amdgcn_wmma_f32_16x16x4_f32(false, av, false, bv1,
                                                         (short)0, acc1, false, false);
        }
        const int mbase = m0 + half * 8;
#pragma unroll
        for (int i = 0; i < 8; ++i) {
            int r = mbase + i;
            if (r < M) {
                if (n0ok) Yb[(long)r * N + nc0] = alpha * acc0[i];
                if (n1ok) Yb[(long)r * N + nc1] = alpha * acc1[i];
            }
        }
    }
}

// ---------------------------------------------------------------------------
// Farthest point sampling. One block per batch; dist[] cached in LDS.
// ---------------------------------------------------------------------------
__global__ void k_fps(const float* __restrict__ xyz, int N, int npoint,
                      int* __restrict__ outIdx)
{
    int b = blockIdx.x, t = threadIdx.x;
    __shared__ float dist[4096];
    __shared__ float rv[256];
    __shared__ int   ri[256];
    __shared__ int   sFar;
    const float* xb = xyz + (long)b * 3 * N;
    for (int j = t; j < N; j += 256) dist[j] = 1e10f;
    if (t == 0) sFar = 0;
    __syncthreads();
    for (int i = 0; i < npoint; ++i) {
        int far = sFar;
        if (t == 0) outIdx[b * npoint + i] = far;
        float cx = xb[far], cy = xb[N + far], cz = xb[2 * N + far];
        float best = -1.f; int bi = 0;
        for (int j = t; j < N; j += 256) {
            float dx = xb[j] - cx, dy = xb[N + j] - cy, dz = xb[2 * N + j] - cz;
            float dd = fminf(dist[j], dx * dx + dy * dy + dz * dz);
            dist[j] = dd;
            if (dd > best) { best = dd; bi = j; }
        }
        rv[t] = best; ri[t] = bi; __syncthreads();
        for (int off = 128; off > 0; off >>= 1) {
            if (t < off) {
                float ov = rv[t + off]; int oi = ri[t + off];
                if (ov > rv[t] || (ov == rv[t] && oi < ri[t])) { rv[t] = ov; ri[t] = oi; }
            }
            __syncthreads();
        }
        if (t == 0) sFar = ri[0];
        __syncthreads();
    }
}

// ---------------------------------------------------------------------------
// kNN: one block per (b, query). Distances in LDS, kk rounds of argmin.
// ---------------------------------------------------------------------------
__global__ void k_knn(const float* __restrict__ Q, const float* __restrict__ R,
                      int Nq, int Nr, int kk, int* __restrict__ idxOut)
{
    int blk = blockIdx.x;
    int b = blk / Nq, q = blk % Nq, t = threadIdx.x;
    __shared__ float d[4096];
    __shared__ float rv[256];
    __shared__ int   ri[256];
    const float* Qb = Q + (long)b * 3 * Nq;
    const float* Rb = R + (long)b * 3 * Nr;
    float qx = Qb[q], qy = Qb[Nq + q], qz = Qb[2 * Nq + q];
    for (int j = t; j < Nr; j += 256) {
        float dx = Rb[j] - qx, dy = Rb[Nr + j] - qy, dz = Rb[2 * Nr + j] - qz;
        d[j] = dx * dx + dy * dy + dz * dz;
    }
    __syncthreads();
    for (int s = 0; s < kk; ++s) {
        float best = 3e38f; int bi = 0x7fffffff;
        for (int j = t; j < Nr; j += 256) {
            float v = d[j];
            if (v < best) { best = v; bi = j; }
        }
        rv[t] = best; ri[t] = bi; __syncthreads();
        for (int off = 128; off > 0; off >>= 1) {
            if (t < off) {
                float ov = rv[t + off]; int oi = ri[t + off];
                if (ov < rv[t] || (ov == rv[t] && oi < ri[t])) { rv[t] = ov; ri[t] = oi; }
            }
            __syncthreads();
        }
        if (t == 0) { idxOut[((long)b * Nq + q) * kk + s] = ri[0]; d[ri[0]] = 3e38f; }
        __syncthreads();
    }
}

// Gather: out[b, coff+c, k] = in[b, c, idx[b,k]]
__global__ void k_gather(const float* __restrict__ in, const int* __restrict__ idx,
                         float* __restrict__ out, int C, int Nin, int K,
                         int outC, int coff, long total)
{
    long g = (long)blockIdx.x * blockDim.x + threadIdx.x;
    if (g >= total) return;
    int k = (int)(g % K);
    int c = (int)((g / K) % C);
    int b = (int)(g / ((long)K * C));
    int id = idx[b * K + k];
    out[((long)b * outC + (coff + c)) * K + k] = in[((long)b * C + c) * Nin + id];
}

// SA input: out[b,c,p,s] = (c<3) ? xyz[b,c,idx]-newxyz[b,c,p] : feats[b,c-3,idx]
__global__ void k_group(const float* __restrict__ xyz, const float* __restrict__ nxyz,
                        const float* __restrict__ feats, const int* __restrict__ idx,
                        float* __restrict__ out, int Nin, int Cf, int P, int S, long total)
{
    long g = (long)blockIdx.x * blockDim.x + threadIdx.x;
    if (g >= total) return;
    int C = 3 + Cf;
    int s = (int)(g % S);
    long r = g / S;
    int p = (int)(r % P); r /= P;
    int c = (int)(r % C);
    int b = (int)(r / C);
    int id = idx[((long)b * P + p) * S + s];
    float v;
    if (c < 3) v = xyz[((long)b * 3 + c) * Nin + id] - nxyz[((long)b * 3 + c) * P + p];
    else       v = feats[((long)b * Cf + (c - 3)) * Nin + id];
    out[g] = v;
}

// Refine input [B, 3+256+257, 64, 16]
__global__ void k_group_refine(const float* __restrict__ pos2, const float* __restrict__ pos1,
                               const float* __restrict__ feat2, const float* __restrict__ feat1,
                               const int* __restrict__ idx, float* __restrict__ out, long total)
{
    const int M = 256, P = 64, S = 16, C2 = 256, C1 = 257, C = 3 + C2 + C1;
    long g = (long)blockIdx.x * blockDim.x + threadIdx.x;
    if (g >= total) return;
    int s = (int)(g % S);
    long r = g / S;
    int p = (int)(r % P); r /= P;
    int c = (int)(r % C);
    int b = (int)(r / C);
    int id = idx[((long)b * P + p) * S + s];
    float v;
    if (c < 3)           v = pos2[((long)b * 3 + c) * M + id] - pos1[((long)b * 3 + c) * P + p];
    else if (c < 3 + C2) v = feat2[((long)b * C2 + (c - 3)) * M + id];
    else                 v = feat1[((long)b * C1 + (c - 3 - C2)) * P + p];
    out[g] = v;
}

// BN stats over (batch, spatial) per channel. One block per channel.
__global__ void k_bn_stats(const float* __restrict__ Y, float* __restrict__ mean,
                           float* __restrict__ var, int C, long Ns)
{
    int c = blockIdx.x, t = threadIdx.x;
    __shared__ float s1[256];
    __shared__ float s2[256];
    long cnt = (long)BB * Ns;
    float a = 0.f, b2 = 0.f;
    for (long i = t; i < cnt; i += 256) {
        long bb = i / Ns, n = i % Ns;
        float v = Y[(bb * C + c) * Ns + n];
        a += v; b2 += v * v;
    }
    s1[t] = a; s2[t] = b2; __syncthreads();
    for (int off = 128; off > 0; off >>= 1) {
        if (t < off) { s1[t] += s1[t + off]; s2[t] += s2[t + off]; }
        __syncthreads();
    }
    if (t == 0) {
        float m = s1[0] / (float)cnt;
        mean[c] = m;
        var[c] = s2[0] / (float)cnt - m * m;
    }
}

__global__ void k_bn_relu(float* __restrict__ Y, const float* __restrict__ gamma,
                          const float* __restrict__ beta, const float* __restrict__ mean,
                          const float* __restrict__ var, int C, long Ns, long total)
{
    long g = (long)blockIdx.x * blockDim.x + threadIdx.x;
    if (g >= total) return;
    int c = (int)((g / Ns) % C);
    float v = Y[g];
    v = gamma[c] * (v - mean[c]) * rsqrtf(var[c] + BN_EPS) + beta[c];
    Y[g] = fmaxf(v, 0.f);
}

// out[b,c,p] = max_s in[b,c,p,s]
__global__ void k_maxpool(const float* __restrict__ in, float* __restrict__ out,
                          int S, long total)
{
    long g = (long)blockIdx.x * blockDim.x + threadIdx.x;
    if (g >= total) return;
    long base = g * S;
    float m = -3e38f;
    for (int s = 0; s < S; ++s) m = fmaxf(m, in[base + s]);
    out[g] = m;
}

// sq[b,m] = sum_c f[b,c,m]^2
__global__ void k_sqsum(const float* __restrict__ f, float* __restrict__ sq,
                        int C, int M, long total)
{
    long g = (long)blockIdx.x * blockDim.x + threadIdx.x;
    if (g >= total) return;
    int m = (int)(g % M);
    int b = (int)(g / M);
    float s = 0.f;
    for (int c = 0; c < C; ++c) {
        float v = f[((long)b * C + c) * M + m];
        s += v * v;
    }
    sq[g] = s;
}

// Row softmax of (raw[b,n,m] - sq2[b,m]); -sq1[n] is row-constant and cancels.
__global__ void k_softmax_row(const float* __restrict__ raw, const float* __restrict__ sq2,
                              float* __restrict__ out)
{
    int row = blockIdx.x;
    int b = row >> 8, n = row & 255, t = threadIdx.x;
    __shared__ float v[256];
    __shared__ float red[256];
    long base = ((long)b * 256 + n) * 256;
    float x = raw[base + t] - sq2[b * 256 + t];
    v[t] = x; red[t] = x; __syncthreads();
    for (int off = 128; off > 0; off >>= 1) {
        if (t < off) red[t] = fmaxf(red[t], red[t + off]);
        __syncthreads();
    }
    float mx = red[0]; __syncthreads();
    float e = expf(v[t] - mx);
    red[t] = e; __syncthreads();
    for (int off = 128; off > 0; off >>= 1) {
        if (t < off) red[t] += red[t + off];
        __syncthreads();
    }
    out[base + t] = e / red[0];
}

// Per batch: row max+argmax of attn, then top-64 extraction.
__global__ void k_corr_topk(const float* __restrict__ attn, int* __restrict__ topkIdx,
                            float* __restrict__ topkScore, int* __restrict__ corrIdx)
{
    int b = blockIdx.x, t = threadIdx.x;
    __shared__ float srcv[256];
    __shared__ int   srci[256];
    __shared__ float rv[256];
    __shared__ int   ri[256];
    long base = ((long)b * 256 + t) * 256;
    float best = -3e38f; int bi = 0;
    for (int m = 0; m < 256; ++m) {
        float v = attn[base + m];
        if (v > best) { best = v; bi = m; }
    }
    srcv[t] = best; srci[t] = bi; __syncthreads();
    for (int r = 0; r < 64; ++r) {
        rv[t] = srcv[t]; ri[t] = t; __syncthreads();
        for (int off = 128; off > 0; off >>= 1) {
            if (t < off) {
                float ov = rv[t + off]; int oi = ri[t + off];
                if (ov > rv[t] || (ov == rv[t] && oi < ri[t])) { rv[t] = ov; ri[t] = oi; }
            }
            __syncthreads();
        }
        if (t == 0) {
            int n = ri[0];
            topkIdx[b * 64 + r] = n;
            topkScore[b * 64 + r] = rv[0];
            corrIdx[b * 64 + r] = srci[n];
            srcv[n] = -3e38f;
        }
        __syncthreads();
    }
}

__global__ void k_set_score(const float* __restrict__ score, float* __restrict__ out, long total)
{
    long g = (long)blockIdx.x * blockDim.x + threadIdx.x;
    if (g >= total) return;
    int b = (int)(g / 64), k = (int)(g % 64);
    out[((long)b * 257 + 256) * 64 + k] = score[g];
}

__global__ void k_add(const float* __restrict__ a, const float* __restrict__ b,
                      float* __restrict__ out, long total)
{
    long g = (long)blockIdx.x * blockDim.x + threadIdx.x;
    if (g >= total) return;
    out[g] = a[g] + b[g];
}

__global__ void k_copy(const float* __restrict__ in, float* __restrict__ out, long total)
{
    long g = (long)blockIdx.x * blockDim.x + threadIdx.x;
    if (g >= total) return;
    out[g] = in[g];
}

// ---------------------------------------------------------------------------
// Host-side orchestration
// ---------------------------------------------------------------------------
struct Layer { const float *W, *g, *b; int M, K; };

static inline unsigned blks(long total) { return (unsigned)((total + 255) / 256); }

static void launch_gemm(const float* A, const float* X, float* Y,
                        int M, int K, int N, int lda, long strideA, int transA,
                        long strideX, long strideY, float alpha, hipStream_t s)
{
    dim3 grid((unsigned)((N + 127) / 128), (unsigned)((M + 15) / 16), BB);
    bool guard = (K & 3) || (M & 15) || (N & 31);
    if (transA) {
        if (guard) k_gemm<true, true ><<<grid, 128, 0, s>>>(A, X, Y, M, K, N, lda, strideA, strideX, strideY, alpha);
        else       k_gemm<false, true><<<grid, 128, 0, s>>>(A, X, Y, M, K, N, lda, strideA, strideX, strideY, alpha);
    } else {
        if (guard) k_gemm<true, false ><<<grid, 128, 0, s>>>(A, X, Y, M, K, N, lda, strideA, strideX, strideY, alpha);
        else       k_gemm<false, false><<<grid, 128, 0, s>>>(A, X, Y, M, K, N, lda, strideA, strideX, strideY, alpha);
    }
}

static void conv_bn_relu(const Layer& L, const float* Xin, float* Yout,
                         long Ns, float* stats, hipStream_t s)
{
    launch_gemm(L.W, Xin, Yout, L.M, L.K, (int)Ns, /*lda*/L.K, /*strideA*/0L, /*transA*/0,
                (long)L.K * Ns, (long)L.M * Ns, 1.0f, s);
    k_bn_stats<<<L.M, 256, 0, s>>>(Yout, stats, stats + L.M, L.M, Ns);
    long tot = (long)BB * L.M * Ns;
    k_bn_relu<<<blks(tot), 256, 0, s>>>(Yout, L.g, L.b, stats, stats + L.M, L.M, Ns, tot);
}

static void set_abstraction(const float* xyz, int Nin, const float* feats, int Cf,
                            int P, int S, const Layer* mlp, int nmlp,
                            const Layer* mlp2, int nmlp2,
                            float* newxyz, float* outfeat,
                            int* fpsIdx, int* knnIdx,
                            float* bufA, float* bufB, float* stats, hipStream_t s)
{
    k_fps<<<BB, 256, 0, s>>>(xyz, Nin, P, fpsIdx);
    { long tot = (long)BB * 3 * P;
      k_gather<<<blks(tot), 256, 0, s>>>(xyz, fpsIdx, newxyz, 3, Nin, P, 3, 0, tot); }
    k_knn<<<BB * P, 256, 0, s>>>(newxyz, xyz, P, Nin, S, knnIdx);
    int Cin = 3 + Cf;
    { long tot = (long)BB * Cin * P * S;
      k_group<<<blks(tot), 256, 0, s>>>(xyz, newxyz, feats, knnIdx, bufA, Nin, Cf, P, S, tot); }
    float* cur = bufA; float* nxt = bufB;
    long Ns = (long)P * S;
    for (int i = 0; i < nmlp; ++i) {
        conv_bn_relu(mlp[i], cur, nxt, Ns, stats, s);
        float* t = cur; cur = nxt; nxt = t;
    }
    float* pooled = (nmlp2 > 0) ? nxt : outfeat;
    { long tot = (long)BB * mlp[nmlp - 1].M * P;
      k_maxpool<<<blks(tot), 256, 0, s>>>(cur, pooled, S, tot); }
    if (nmlp2 > 0) {
        float* c2 = pooled;
        for (int i = 0; i < nmlp2; ++i) {
            float* dst = (i == nmlp2 - 1) ? outfeat : ((c2 == bufA) ? bufB : bufA);
            conv_bn_relu(mlp2[i], c2, dst, (long)P, stats, s);
            c2 = dst;
        }
    }
}

static void conv_stack(const float* in, const Layer* ls, int nl, float* out,
                       float* bufA, float* bufB, float* stats, hipStream_t s)
{
    const float* cur = in;
    float* nxt = bufA;
    for (int i = 0; i < nl; ++i) {
        float* dst = (i == nl - 1) ? out : nxt;
        conv_bn_relu(ls[i], cur, dst, 256L, stats, s);
        cur = dst;
        nxt = (dst == bufA) ? bufB : bufA;
    }
}

static inline float* wsF(char*& p, size_t n) {
    float* r = (float*)p; p += ((n + 3) & ~size_t(3)) * sizeof(float); return r;
}
static inline int* wsI(char*& p, size_t n) {
    int* r = (int*)p; p += ((n + 3) & ~size_t(3)) * sizeof(int); return r;
}

// ---------------------------------------------------------------------------
extern "C" void kernel_launch(void* const* d_in, const int* in_sizes, int n_in,
                              void* d_out, int out_size, void* d_ws, size_t ws_size,
                              hipStream_t stream)
{
    (void)in_sizes; (void)n_in; (void)out_size; (void)ws_size;
    const float* pc1 = (const float*)d_in[0];
    const float* pc2 = (const float*)d_in[1];
    // feature1/feature2 (d_in[2], d_in[3]) unused: sa1 in_channel==0

    int pi = 4;
    auto L = [&](int M, int K) -> Layer {
        Layer l; l.W = (const float*)d_in[pi]; l.g = (const float*)d_in[pi + 1];
        l.b = (const float*)d_in[pi + 2]; pi += 3; l.M = M; l.K = K; return l;
    };
    Layer sa1m[3]  = { L(64, 3),   L(64, 64),   L(128, 64) };
    Layer sa1m2[1] = { L(128, 128) };
    Layer sa2m[3]  = { L(128, 131), L(128, 128), L(256, 128) };
    // sa2_mlp2 is empty
    Layer convs[3] = { L(256, 256), L(256, 256), L(512, 256) };
    Layer fem[2]   = { L(512, 516), L(256, 512) };
    Layer fem2[3]  = { L(256, 256), L(128, 256), L(3, 128) };

    // ---- workspace layout (bump allocated) ----
    char* wp = (char*)d_ws;
    float* bufA = wsF(wp, (size_t)BB * 131 * 16384);   // max layer input  (~34.3M f)
    float* bufB = wsF(wp, (size_t)BB * 256 * 16384);   // max layer output (~67.1M f)
    float* l1_pc1 = wsF(wp, (size_t)BB * 3 * 512);
    float* l1_f1  = wsF(wp, (size_t)BB * 128 * 512);
    float* l1_pc2 = wsF(wp, (size_t)BB * 3 * 512);
    float* l1_f2  = wsF(wp, (size_t)BB * 128 * 512);
    float* l2_pc1 = wsF(wp, (size_t)BB * 3 * 256);
    float* l2_f1  = wsF(wp, (size_t)BB * 256 * 256);
    float* l2_pc2 = wsF(wp, (size_t)BB * 3 * 256);
    float* l2_f2  = wsF(wp, (size_t)BB * 256 * 256);
    float* f1n    = wsF(wp, (size_t)BB * 512 * 256);
    float* f2n    = wsF(wp, (size_t)BB * 512 * 256);
    float* sq2    = wsF(wp, (size_t)BB * 256);
    float* stats  = wsF(wp, 1024 + 16);
    float* srcCorrPc = wsF(wp, (size_t)BB * 3 * 64);
    float* feat1Ref  = wsF(wp, (size_t)BB * 257 * 64);
    float* topkScore = wsF(wp, (size_t)BB * 64);
    int* fpsIdx  = wsI(wp, (size_t)BB * 512);
    int* knnIdx  = wsI(wp, (size_t)BB * 512 * 32);     // reused: sa2 needs 16*256*64 (same)
    int* feIdx   = wsI(wp, (size_t)BB * 64 * 16);
    int* topkIdx = wsI(wp, (size_t)BB * 64);
    int* corrIdxK = wsI(wp, (size_t)BB * 64);

    // ---- two set-abstraction stages per cloud ----
    set_abstraction(pc1, 4096, nullptr, 0, 512, 32, sa1m, 3, sa1m2, 1,
                    l1_pc1, l1_f1, fpsIdx, knnIdx, bufA, bufB, stats, stream);
    set_abstraction(l1_pc1, 512, l1_f1, 128, 256, 64, sa2m, 3, nullptr, 0,
                    l2_pc1, l2_f1, fpsIdx, knnIdx, bufA, bufB, stats, stream);
    set_abstraction(pc2, 4096, nullptr, 0, 512, 32, sa1m, 3, sa1m2, 1,
                    l1_pc2, l1_f2, fpsIdx, knnIdx, bufA, bufB, stats, stream);
    set_abstraction(l1_pc2, 512, l1_f2, 128, 256, 64, sa2m, 3, nullptr, 0,
                    l2_pc2, l2_f2, fpsIdx, knnIdx, bufA, bufB, stats, stream);

    // ---- conv stacks ----
    conv_stack(l2_f1, convs, 3, f1n, bufA, bufB, stats, stream);
    conv_stack(l2_f2, convs, 3, f2n, bufA, bufB, stats, stream);

    // ---- attention: softmax(2*f1n^T f2n - sq2[m]) ----
    { long tot = (long)BB * 256;
      k_sqsum<<<blks(tot), 256, 0, stream>>>(f2n, sq2, 512, 256, tot); }
    launch_gemm(f1n, f2n, bufA, 256, 512, 256, /*lda*/256, /*strideA*/(long)512 * 256,
                /*transA*/1, (long)512 * 256, (long)256 * 256, 2.0f, stream);
    float* outAttn = (float*)d_out + 30720;
    k_softmax_row<<<BB * 256, 256, 0, stream>>>(bufA, sq2, outAttn);

    // ---- correspondence top-64 ----
    k_corr_topk<<<BB, 256, 0, stream>>>(outAttn, topkIdx, topkScore, corrIdxK);
    { long tot = (long)BB * 256 * 64;   // src_corr_feature -> feat1Ref channels [0,256)
      k_gather<<<blks(tot), 256, 0, stream>>>(l2_f1, topkIdx, feat1Ref, 256, 256, 64, 257, 0, tot); }
    { long tot = (long)BB * 64;         // feat1Ref channel 256 = topk_score
      k_set_score<<<blks(tot), 256, 0, stream>>>(topkScore, feat1Ref, tot); }
    { long tot = (long)BB * 3 * 64;     // src_corr_pc = l2_pc2[:, src_corr_idx_k]
      k_gather<<<blks(tot), 256, 0, stream>>>(l2_pc2, corrIdxK, srcCorrPc, 3, 256, 64, 3, 0, tot); }
    { long tot = (long)BB * 3 * 64;     // src_interest -> d_out region 0
      k_gather<<<blks(tot), 256, 0, stream>>>(l2_pc1, topkIdx, (float*)d_out, 3, 256, 64, 3, 0, tot); }

    // ---- position refinement ----
    k_knn<<<BB * 64, 256, 0, stream>>>(srcCorrPc, l2_pc2, 64, 256, 16, feIdx);
    { long tot = (long)BB * 516 * 64 * 16;
      k_group_refine<<<blks(tot), 256, 0, stream>>>(l2_pc2, srcCorrPc, l2_f2, feat1Ref,
                                                    feIdx, bufB, tot); }
    conv_bn_relu(fem[0], bufB, bufA, 1024L, stats, stream);   // 516 -> 512
    conv_bn_relu(fem[1], bufA, bufB, 1024L, stats, stream);   // 512 -> 256
    { long tot = (long)BB * 256 * 64;                         // max over nsample=16
      k_maxpool<<<blks(tot), 256, 0, stream>>>(bufB, bufA, 16, tot); }
    conv_bn_relu(fem2[0], bufA, bufB, 64L, stats, stream);    // 256 -> 256
    conv_bn_relu(fem2[1], bufB, bufA, 64L, stats, stream);    // 256 -> 128
    conv_bn_relu(fem2[2], bufA, bufB, 64L, stats, stream);    // 128 -> 3  (BN+ReLU, faithful)

    // ---- final outputs ----
    { long tot = (long)BB * 3 * 64;     // src_corr_pc + pos_diff -> region 1
      k_add<<<blks(tot), 256, 0, stream>>>(srcCorrPc, bufB, (float*)d_out + 3072, tot); }
    { long tot = (long)BB * 3 * 256;
      k_copy<<<blks(tot), 256, 0, stream>>>(l2_pc1, (float*)d_out + 6144, tot); }
    { long tot = (long)BB * 3 * 256;
      k_copy<<<blks(tot), 256, 0, stream>>>(l2_pc2, (float*)d_out + 18432, tot); }
}